// NFFT_86535001080598
// MI455X (gfx1250) — compile-verified
//
#include <hip/hip_runtime.h>
#include <math.h>

// ---------------------------------------------------------------------------
// NFFT forward for B=2, P=524288, N=(512,512), NGRID=(1024,1024), M=4, sigma=2
//   beta = pi*(2 - 1/2) = 1.5*pi  (both dims)
// Pipeline: deconvolve+pad -> 1024-pt WMMA FFT rows -> cols -> 8x8 gather
// ---------------------------------------------------------------------------

typedef float v2f __attribute__((ext_vector_type(2)));
typedef float v8f __attribute__((ext_vector_type(8)));

#define NG     1024
#define NHALF  512
#define MWIN   4
#define PI_F   3.14159265358979323846f
#define BETA_F 4.71238898038468986f   /* 1.5*pi */
#define YPAD   34                     /* padded row stride (float2) for LDS Y buffer */

__device__ __forceinline__ v8f wmma4(v2f a, v2f b, v8f c) {
  // D(16x16 f32) = A(16x4) @ B(4x16) + C ; f32 WMMA has no A/B negation (ISA 7.12)
  return __builtin_amdgcn_wmma_f32_16x16x4_f32(false, a, false, b, (short)0, c,
                                               false, false);
}

// e^{-2*pi*i*idx/m} via fast hardware sin/cos (arguments are pre-reduced ints)
__device__ __forceinline__ void twiddle(int idx, float invm_times_2pi,
                                        float* s, float* c) {
  float a = -(float)idx * invm_times_2pi;
  __sincosf(a, s, c);
}

// ---------------------------------------------------------------------------
// Kernel 1: deconvolve f_hat by window FT, scale by 1/n^2, write zero-padded
// ifftshift-ed grid:  grid[b][(k1 mod 1024)][(k2 mod 1024)] = fh[k]/(ft*n^2)
// ---------------------------------------------------------------------------
__device__ __forceinline__ float i0_large(float x) {
  // Asymptotic series, valid here (x in [17.7, 18.9]); ~1e-7 relative error.
  float t = 1.0f / x;
  float s = 1.0f + t * (0.125f + t * (0.0703125f +
              t * (0.0732421875f + t * 0.112152099609375f)));
  return expf(x) * rsqrtf(2.0f * PI_F * x) * s;
}

__device__ __forceinline__ float phi_hat(int k) {
  float w   = 2.0f * PI_F * (float)k / (float)NG;
  float arg = sqrtf(BETA_F * BETA_F - w * w);
  return i0_large((float)MWIN * arg) * (1.0f / (float)NG);
}

__global__ void nfft_deconv(const float* __restrict__ fh,
                            float2* __restrict__ grid, int B) {
  int gid   = blockIdx.x * blockDim.x + threadIdx.x;
  int total = B * NG * NG;
  if (gid >= total) return;
  int b   = gid >> 20;               // /(1024*1024)
  int rem = gid & (NG * NG - 1);
  int p1  = rem >> 10;
  int p2  = rem & (NG - 1);
  float2 val = make_float2(0.0f, 0.0f);
  bool z1 = (p1 >= 256 && p1 < 768);
  bool z2 = (p2 >= 256 && p2 < 768);
  if (!z1 && !z2) {
    int k1 = (p1 >= 768) ? p1 - NG : p1;   // k in [-256,256)
    int k2 = (p2 >= 768) ? p2 - NG : p2;
    int i1 = k1 + 256, i2 = k2 + 256;      // index into f_hat
    float ft    = phi_hat(k1) * phi_hat(k2);
    float scale = 1.0f / (ft * (float)NG * (float)NG);
    const float* src = fh + (((size_t)b * NHALF + i1) * NHALF + i2) * 2;
    val.x = src[0] * scale;
    val.y = src[1] * scale;
  }
  grid[gid] = val;
}

// ---------------------------------------------------------------------------
// Kernel 2: in-place 1024-point forward FFT (e^{-2pi i}) along a line, one
// wave per line. Four-step 32x32 decomposition; both 32x32 complex matmuls
// run on v_wmma_f32_16x16x4_f32 (2x2 tiles x 8 K-chunks x 4 real matmuls).
// Branch-free body: EXEC is all-1s at every WMMA (ISA 7.12 requirement).
// elemStride = 1 (rows) or 1024 (columns).
// ---------------------------------------------------------------------------
__global__ void __launch_bounds__(32)
fft1024_wmma(float2* __restrict__ g, int elemStride) {
  __shared__ __align__(16) float2 Xl[NG];          // input line, linear
  __shared__ __align__(16) float2 Yt[32 * YPAD];   // Y transposed: [k2][n1], padded

  const int lane = threadIdx.x;
  const int hi   = lane >> 4;    // half-wave: selects K subset {2,3} / rows {2,3}
  const int ln   = lane & 15;

  const float W32A  = 2.0f * PI_F / 32.0f;
  const float W1024 = 2.0f * PI_F / 1024.0f;

  const int batch = blockIdx.x >> 10;
  const int line  = blockIdx.x & 1023;
  const size_t base = (size_t)batch * (NG * NG) +
                      (elemStride == 1 ? (size_t)line * NG : (size_t)line);

  // ---- load line (coalesced for rows; column pass is L2-resident) ----
  for (int it = 0; it < 32; ++it) {
    int idx = it * 32 + lane;
    Xl[idx] = g[base + (size_t)idx * (size_t)elemStride];
  }
  __syncthreads();

  // ---- Step A: Y[n1,k2] = sum_{n2} Xm[n1,n2] * F32[n2,k2], Xm[n1,n2]=Xl[n1+32*n2]
  for (int nt = 0; nt < 2; ++nt) {
    v8f yre[2], yim[2];
    #pragma unroll
    for (int mt = 0; mt < 2; ++mt) {
      yre[mt] = (v8f){0,0,0,0,0,0,0,0};
      yim[mt] = (v8f){0,0,0,0,0,0,0,0};
    }
    #pragma unroll
    for (int c = 0; c < 8; ++c) {
      const int r0  = 4 * c + 2 * hi;        // K rows handled by this half-wave
      const int col = 16 * nt + ln;          // N index (k2)
      float s0, c0, s1, c1;                  // B = F32 chunk, F32[r,c]=e^{-2pi i r c/32}
      twiddle((r0 * col) & 31, W32A, &s0, &c0);
      twiddle(((r0 + 1) * col) & 31, W32A, &s1, &c1);
      const v2f bre  = {c0, c1};
      const v2f bim  = {s0, s1};
      const v2f nbim = {-s0, -s1};
      #pragma unroll
      for (int mt = 0; mt < 2; ++mt) {
        const int row = 16 * mt + ln;        // M index (n1)
        float2 e0 = Xl[row + 32 * r0];       // Xm[row, r0]
        float2 e1 = Xl[row + 32 * (r0 + 1)]; // Xm[row, r0+1]
        v2f are = {e0.x, e1.x};
        v2f aim = {e0.y, e1.y};
        yre[mt] = wmma4(are, bre,  yre[mt]); // Re += Are*Bre
        yre[mt] = wmma4(aim, nbim, yre[mt]); // Re -= Aim*Bim
        yim[mt] = wmma4(are, bim,  yim[mt]); // Im += Are*Bim
        yim[mt] = wmma4(aim, bre,  yim[mt]); // Im += Aim*Bre
      }
    }
    // twiddle W1024^{n1*k2} in-register on D fragments, store transposed
    #pragma unroll
    for (int mt = 0; mt < 2; ++mt) {
      #pragma unroll
      for (int v = 0; v < 8; ++v) {
        const int row = 16 * mt + v + 8 * hi;   // n1 (D layout: rows v / v+8)
        const int col = 16 * nt + ln;           // k2
        float st, ct;
        twiddle(row * col, W1024, &st, &ct);    // row*col <= 961 < 1024
        float re = yre[mt][v], im = yim[mt][v];
        Yt[col * YPAD + row] = make_float2(re * ct - im * st,
                                           re * st + im * ct);
      }
    }
  }
  __syncthreads();

  // ---- Step C: Z[k1,k2] = sum_{n1} F32[k1,n1] * Y[n1,k2]; out[32*k1+k2]=Z ----
  for (int nt = 0; nt < 2; ++nt) {
    v8f zre[2], zim[2];
    #pragma unroll
    for (int mt = 0; mt < 2; ++mt) {
      zre[mt] = (v8f){0,0,0,0,0,0,0,0};
      zim[mt] = (v8f){0,0,0,0,0,0,0,0};
    }
    #pragma unroll
    for (int c = 0; c < 8; ++c) {
      const int r0  = 4 * c + 2 * hi;
      const int col = 16 * nt + ln;
      // B chunk from Yt: rows r0, r0+1 at column col; 16B-aligned ds_read_b128
      const float4 raw = *(const float4*)&Yt[col * YPAD + r0];
      const v2f bre = {raw.x, raw.z};
      const v2f bim = {raw.y, raw.w};
      #pragma unroll
      for (int mt = 0; mt < 2; ++mt) {
        const int row = 16 * mt + ln;         // k1
        float s0, c0, s1, c1;                 // A = F32 chunk
        twiddle((row * r0) & 31, W32A, &s0, &c0);
        twiddle((row * (r0 + 1)) & 31, W32A, &s1, &c1);
        const v2f are  = {c0, c1};
        const v2f aim  = {s0, s1};
        const v2f naim = {-s0, -s1};
        zre[mt] = wmma4(are,  bre, zre[mt]);
        zre[mt] = wmma4(naim, bim, zre[mt]);
        zim[mt] = wmma4(are,  bim, zim[mt]);
        zim[mt] = wmma4(aim,  bre, zim[mt]);
      }
    }
    #pragma unroll
    for (int mt = 0; mt < 2; ++mt) {
      #pragma unroll
      for (int v = 0; v < 8; ++v) {
        const int row  = 16 * mt + v + 8 * hi;  // k1
        const int col  = 16 * nt + ln;          // k2
        const int oidx = 32 * row + col;
        g[base + (size_t)oidx * (size_t)elemStride] =
            make_float2(zre[mt][v], zim[mt][v]);
      }
    }
  }
}

// ---------------------------------------------------------------------------
// Kernel 3: per-point 8x8 Kaiser-Bessel gather (grid is L2-resident: 16 MB).
// Fast-path math: __expf + v_rcp (z in [0, 18.85], e^z >= 1, no edge cases).
// ---------------------------------------------------------------------------
__device__ __forceinline__ float kb_weight(float f) {
  float u2 = (float)(MWIN * MWIN) - f * f;
  float u  = sqrtf(fmaxf(u2, 1e-12f));
  float z  = BETA_F * u;
  float ez = __expf(z);
  float sh = 0.5f * (ez - __builtin_amdgcn_rcpf(ez));
  float w  = sh * __builtin_amdgcn_rcpf(PI_F * u);
  return (u2 > 0.0f) ? w : 0.0f;
}

__global__ void nfft_gather(const float* __restrict__ x,
                            const float2* __restrict__ grid,
                            float* __restrict__ out, int BP, int P) {
  int j = blockIdx.x * blockDim.x + threadIdx.x;
  if (j >= BP) return;
  int b = j / P;
  float x1 = x[2 * j], x2 = x[2 * j + 1];

  float w1[8], w2[8];
  int id1[8], id2[8];
  int l01 = (int)floorf(x1 * (float)NG);
  int l02 = (int)floorf(x2 * (float)NG);
  #pragma unroll
  for (int i = 0; i < 8; ++i) {
    int l = l01 - (MWIN - 1) + i;                  // offs = -3..4
    w1[i]  = kb_weight(x1 * (float)NG - (float)l);
    id1[i] = l & (NG - 1);

    l = l02 - (MWIN - 1) + i;
    w2[i]  = kb_weight(x2 * (float)NG - (float)l);
    id2[i] = l & (NG - 1);
  }

  const float2* gb = grid + (size_t)b * NG * NG;
  float accr = 0.0f, acci = 0.0f;
  for (int i = 0; i < 8; ++i) {
    const float2* rowp = gb + (size_t)id1[i] * NG;
    float wi = w1[i];
    #pragma unroll
    for (int jj = 0; jj < 8; ++jj) {
      float2 gv = rowp[id2[jj]];
      float  w  = wi * w2[jj];
      accr += w * gv.x;
      acci += w * gv.y;
    }
  }
  out[2 * j]     = accr;
  out[2 * j + 1] = acci;
}

// ---------------------------------------------------------------------------
extern "C" void kernel_launch(void* const* d_in, const int* in_sizes, int n_in,
                              void* d_out, int out_size, void* d_ws,
                              size_t ws_size, hipStream_t stream) {
  const float* x  = (const float*)d_in[0];   // [B,P,2]
  const float* fh = (const float*)d_in[1];   // [B,512,512,2]
  float* out      = (float*)d_out;           // [B,P,2]
  float2* grid    = (float2*)d_ws;           // B*1024*1024 float2 = 16 MB scratch

  const int BP = in_sizes[0] / 2;
  const int B  = in_sizes[1] / (NHALF * NHALF * 2);
  const int P  = BP / B;

  const int cells = B * NG * NG;
  nfft_deconv<<<(cells + 255) / 256, 256, 0, stream>>>(fh, grid, B);
  fft1024_wmma<<<B * NG, 32, 0, stream>>>(grid, 1);    // row FFTs
  fft1024_wmma<<<B * NG, 32, 0, stream>>>(grid, NG);   // column FFTs
  nfft_gather<<<(BP + 255) / 256, 256, 0, stream>>>(x, grid, out, BP, P);
}